// DenseGCM_33182917329467
// MI455X (gfx1250) — compile-verified
//
#include <hip/hip_runtime.h>
#include <hip/hip_bf16.h>

typedef __attribute__((ext_vector_type(2))) float v2f;
typedef __attribute__((ext_vector_type(4))) float v4f;
typedef __attribute__((ext_vector_type(8))) float v8f;

#define B_ 64
#define N_ 1024
#define F_ 64

// ---------------------------------------------------------------------------
// Kernel 1: per-batch meta. r = effective num_nodes (after possible decrement),
// of = overflow flag, and the num_nodes+1 output (written as float).
// ---------------------------------------------------------------------------
__global__ void meta_kernel(const long long* __restrict__ num_nodes,
                            int* __restrict__ r_of,      // [0..63]=r, [64..127]=of
                            float* __restrict__ out_nn)  // 64 floats
{
    int b = threadIdx.x;
    if (b >= B_) return;
    long long nn = num_nodes[b];
    int of = (nn + 1 > (long long)N_) ? 1 : 0;
    int r  = of ? (int)nn - 1 : (int)nn;
    r_of[b]       = r;
    r_of[B_ + b]  = of;
    // reference returns (shifted num_nodes) + 1 == r + 1 in both branches
    out_nn[b] = (float)(r + 1);
}

// ---------------------------------------------------------------------------
// Kernel 2: streaming copy of adj / weights with the (statically dead but
// implemented) overflow shift. One block per (b,i) row; 256 threads x 16B
// covers the 4KB row. Non-temporal: 512MB stream, don't pollute L2.
// ---------------------------------------------------------------------------
__global__ void copy_shift_kernel(const float* __restrict__ src,
                                  float* __restrict__ dst,
                                  const int* __restrict__ r_of)
{
    int row = blockIdx.x;          // b*N + i
    int b = row >> 10;
    int i = row & (N_ - 1);
    int of = r_of[B_ + b];
    int t  = threadIdx.x;          // 0..255
    size_t base = (size_t)row * N_;
    if (!of) {
        const v4f* s = (const v4f*)(src + base) + t;
        v4f*       d = (v4f*)(dst + base) + t;
        v4f v = __builtin_nontemporal_load(s);
        __builtin_nontemporal_store(v, d);
    } else {
        // rolled (-1,-1) with row0/col0 zeroed: dst[b,i,j] =
        //   (i==N-1 || j==N-1) ? 0 : src[b,i+1,j+1]
        size_t sbase = ((size_t)b * N_ + (i + 1)) * N_;
        v4f v;
        #pragma unroll
        for (int k = 0; k < 4; ++k) {
            int j = t * 4 + k;
            v[k] = (i == N_ - 1 || j == N_ - 1) ? 0.0f : src[sbase + j + 1];
        }
        __builtin_nontemporal_store(v, (v4f*)(dst + base) + t);
    }
}

// ---------------------------------------------------------------------------
// Kernel 3: one block per batch. Builds nodes_in on the fly (shift + x insert),
// writes nodes_out = nodes_in + masked sinusoidal embedding, and accumulates
// AGG[b,f] = sum_j adj_eff[b, r, j] * nodes_in[b, j, f]  (the only einsum row
// that is observable through mx). adj row staged in LDS.
// ---------------------------------------------------------------------------
__global__ void nodes_agg_kernel(const float* __restrict__ x,
                                 const float* __restrict__ nodes,
                                 const float* __restrict__ adj,
                                 const int* __restrict__ r_of,
                                 float* __restrict__ nodes_out,
                                 float* __restrict__ AGG)
{
    __shared__ float adjrow[N_];
    __shared__ float accred[256];

    int b   = blockIdx.x;
    int tid = threadIdx.x;           // 0..255
    int r   = r_of[b];
    int of  = r_of[B_ + b];

    // stage adj_eff[b, r, :] into LDS
    size_t arow = ((size_t)b * N_ + r) * N_;
    size_t arow_s = ((size_t)b * N_ + (r + 1)) * N_;   // shifted source row
    for (int j = tid; j < N_; j += 256) {
        float av;
        if (!of)                       av = adj[arow + j];
        else if (r == N_ - 1 || j == N_ - 1) av = 0.0f;
        else                           av = adj[arow_s + j + 1];
        adjrow[j] = av;
    }
    __syncthreads();

    int f   = tid & (F_ - 1);
    int sub = tid >> 6;              // 0..3
    // inv_freq[f] = 10000^(-f/512) = exp2(-f * log2(10000)/512)
    float invf = exp2f(-(float)f * (13.287712379549449f / 512.0f));
    float xv   = x[(size_t)b * F_ + f];

    float acc = 0.0f;
    for (int p = sub; p < N_; p += 4) {
        float v;
        if (p == r)        v = xv;
        else if (!of)      v = nodes[((size_t)b * N_ + p) * F_ + f];
        else if (p == N_-1) v = 0.0f;
        else               v = nodes[((size_t)b * N_ + p + 1) * F_ + f];
        acc = fmaf(adjrow[p], v, acc);
        float e = (p <= r) ? sinf((float)p * invf) : 0.0f;
        nodes_out[((size_t)b * N_ + p) * F_ + f] = v + e;
    }

    accred[tid] = acc;
    __syncthreads();
    if (tid < F_) {
        AGG[(size_t)b * F_ + tid] =
            accred[tid] + accred[64 + tid] + accred[128 + tid] + accred[192 + tid];
    }
}

// ---------------------------------------------------------------------------
// Kernel 4: mx = tanh(AGG @ W), a 64x64x64 f32 GEMM on the tensor path.
// One wave per 16x16 output tile, V_WMMA_F32_16X16X4_F32, K accumulated in
// 16 steps of 4. Exact f32 — matches the f32 reference precision.
// VGPR layouts per ISA 7.12.2:
//   A 16x4 : lane L -> M = L%16, K pair = (L/16)*2 .. +1  (v2f)
//   B 4x16 : lane L -> N = L%16, K pair = (L/16)*2 .. +1  (v2f)
//   C/D    : VGPR v -> M = v + 8*(L>=16), N = L%16        (v8f)
// ---------------------------------------------------------------------------
__global__ void mx_wmma_kernel(const float* __restrict__ AGG,  // 64x64
                               const float* __restrict__ W,    // 64x64
                               float* __restrict__ mx)         // 64x64
{
    int tile = blockIdx.x;           // 0..15
    int tm = tile >> 2;
    int tn = tile & 3;
    int lane = threadIdx.x;          // 0..31, EXEC all ones (no divergence)
    int half = lane >> 4;            // selects K pair
    int l16  = lane & 15;

    v8f c = {};
    #pragma unroll
    for (int k0 = 0; k0 < 64; k0 += 4) {
        int kb = k0 + half * 2;
        const float* ap = AGG + (tm * 16 + l16) * 64 + kb;
        v2f a; a.x = ap[0];  a.y = ap[1];
        const float* bp = W + kb * 64 + tn * 16 + l16;
        v2f bfrag; bfrag.x = bp[0]; bfrag.y = bp[64];
        c = __builtin_amdgcn_wmma_f32_16x16x4_f32(
                /*neg_a=*/false, a, /*neg_b=*/false, bfrag,
                /*c_mod=*/(short)0, c, /*reuse_a=*/false, /*reuse_b=*/false);
    }

    #pragma unroll
    for (int v = 0; v < 8; ++v) {
        int row = tm * 16 + v + half * 8;   // batch index b
        int col = tn * 16 + l16;            // feature f
        mx[row * 64 + col] = tanhf(c[v]);
    }
}

// ---------------------------------------------------------------------------
extern "C" void kernel_launch(void* const* d_in, const int* in_sizes, int n_in,
                              void* d_out, int out_size, void* d_ws, size_t ws_size,
                              hipStream_t stream)
{
    const float*     x         = (const float*)d_in[0];   // 64*64
    const float*     nodes     = (const float*)d_in[1];   // 64*1024*64
    const float*     adj       = (const float*)d_in[2];   // 64*1024*1024
    const float*     weights   = (const float*)d_in[3];   // 64*1024*1024
    const float*     W         = (const float*)d_in[4];   // 64*64
    const long long* num_nodes = (const long long*)d_in[5];

    float* out = (float*)d_out;
    const size_t n_mx    = (size_t)B_ * F_;            // 4096
    const size_t n_nodes = (size_t)B_ * N_ * F_;       // 4,194,304
    const size_t n_mat   = (size_t)B_ * N_ * N_;       // 67,108,864
    float* out_mx      = out;
    float* out_nodes   = out + n_mx;
    float* out_adj     = out + n_mx + n_nodes;
    float* out_weights = out + n_mx + n_nodes + n_mat;
    float* out_nn      = out + n_mx + n_nodes + 2 * n_mat;

    int*   r_of = (int*)d_ws;                                  // 128 ints
    float* AGG  = (float*)((char*)d_ws + 512);                 // 64*64 floats

    meta_kernel<<<1, 64, 0, stream>>>(num_nodes, r_of, out_nn);

    nodes_agg_kernel<<<B_, 256, 0, stream>>>(x, nodes, adj, r_of, out_nodes, AGG);

    copy_shift_kernel<<<B_ * N_, 256, 0, stream>>>(adj,     out_adj,     r_of);
    copy_shift_kernel<<<B_ * N_, 256, 0, stream>>>(weights, out_weights, r_of);

    mx_wmma_kernel<<<16, 32, 0, stream>>>(AGG, W, out_mx);
}